// GMLPModel_66846870995045
// MI455X (gfx1250) — compile-verified
//
#include <hip/hip_runtime.h>

typedef float v2f __attribute__((ext_vector_type(2)));
typedef float v8f __attribute__((ext_vector_type(8)));

#define B_ROWS 8192
#define IN_DIM 1024
#define HID    512
#define OUT_D  128
#define EPS_LN  1e-6f
#define EPS_COS 1e-8f

static __device__ __forceinline__ v8f wmma_f32(v2f a, v2f b, v8f c) {
    // V_WMMA_F32_16X16X4_F32: D = A(16x4) * B(4x16) + C(16x16), all fp32
    return __builtin_amdgcn_wmma_f32_16x16x4_f32(
        /*neg_a=*/false, a, /*neg_b=*/false, b,
        /*c_mod=*/(short)0, c, /*reuse_a=*/false, /*reuse_b=*/false);
}

// ---------------------------------------------------------------------------
// Kernel 1: h = LayerNorm(PReLU(x @ w1 + b1))   [8192 x 512]
// One block = 32 rows (2 row-tiles/wave); 8 waves; wave w owns cols
// [w*64, w*64+64) (4 WMMA tiles). Each B fetch feeds 2 WMMAs. LayerNorm row
// stats via half-wave shfl_xor reductions + 2KB LDS cross-wave exchange;
// normalized output written straight from the accumulators (no tile staging).
// ---------------------------------------------------------------------------
__global__ __launch_bounds__(256) void k_mlp1(
    const float* __restrict__ x, const float* __restrict__ w1,
    const float* __restrict__ b1, const float* __restrict__ pa,
    const float* __restrict__ gamma, const float* __restrict__ beta,
    float* __restrict__ hn)
{
    __shared__ float ps[8][32];
    __shared__ float ps2[8][32];
    __shared__ float smean[32];
    __shared__ float srstd[32];

    const int tid  = threadIdx.x;
    const int wave = tid >> 5;
    const int lane = tid & 31;
    const int half = lane >> 4;              // K sub-pair select in A/B layout
    const int l16  = lane & 15;              // M for A-lanes, N for B-lanes
    const int rowBase = blockIdx.x * 32;

    const float* xr0 = x + (size_t)(rowBase + l16) * IN_DIM;
    const float* xr1 = xr0 + (size_t)16 * IN_DIM;

    v8f acc[2][4] = {{v8f{0.f}, v8f{0.f}, v8f{0.f}, v8f{0.f}},
                     {v8f{0.f}, v8f{0.f}, v8f{0.f}, v8f{0.f}}};
    for (int k0 = 0; k0 < IN_DIM; k0 += 4) {
        const int ka = k0 + half * 2;
        const v2f a0 = *(const v2f*)(xr0 + ka);
        const v2f a1 = *(const v2f*)(xr1 + ka);
        const float* w1r = w1 + (size_t)ka * HID;
#pragma unroll
        for (int t = 0; t < 4; ++t) {
            const int n = wave * 64 + t * 16 + l16;
            v2f b;
            b.x = w1r[n];
            b.y = w1r[HID + n];
            acc[0][t] = wmma_f32(a0, b, acc[0][t]);
            acc[1][t] = wmma_f32(a1, b, acc[1][t]);
        }
    }

    // bias + PReLU in-register, accumulate per-lane LN partial sums
    const float alpha = pa[0];
    float s1[2][8] = {};
    float s2[2][8] = {};
#pragma unroll
    for (int rt = 0; rt < 2; ++rt) {
#pragma unroll
        for (int t = 0; t < 4; ++t) {
            const float bias = b1[wave * 64 + t * 16 + l16];
#pragma unroll
            for (int r = 0; r < 8; ++r) {
                float v = acc[rt][t][r] + bias;
                v = (v >= 0.0f) ? v : alpha * v;       // PReLU
                acc[rt][t][r] = v;
                s1[rt][r] += v;
                s2[rt][r] += v * v;
            }
        }
    }
    // reduce over the 16 lanes of each half-wave (rows live per-half)
#pragma unroll
    for (int rt = 0; rt < 2; ++rt) {
#pragma unroll
        for (int r = 0; r < 8; ++r) {
#pragma unroll
            for (int m = 8; m >= 1; m >>= 1) {
                s1[rt][r] += __shfl_xor(s1[rt][r], m, 16);
                s2[rt][r] += __shfl_xor(s2[rt][r], m, 16);
            }
        }
    }
    if (l16 == 0) {
#pragma unroll
        for (int rt = 0; rt < 2; ++rt) {
#pragma unroll
            for (int r = 0; r < 8; ++r) {
                const int row = rt * 16 + half * 8 + r;
                ps[wave][row]  = s1[rt][r];
                ps2[wave][row] = s2[rt][r];
            }
        }
    }
    __syncthreads();
    if (tid < 32) {
        float a = 0.f, b = 0.f;
#pragma unroll
        for (int w = 0; w < 8; ++w) { a += ps[w][tid]; b += ps2[w][tid]; }
        const float mean = a * (1.0f / HID);
        const float var  = b * (1.0f / HID) - mean * mean;
        smean[tid] = mean;
        srstd[tid] = rsqrtf(var + EPS_LN);
    }
    __syncthreads();

    float mu[2][8], rs[2][8];
#pragma unroll
    for (int rt = 0; rt < 2; ++rt) {
#pragma unroll
        for (int r = 0; r < 8; ++r) {
            const int row = rt * 16 + half * 8 + r;
            mu[rt][r] = smean[row];
            rs[rt][r] = srstd[row];
        }
    }
#pragma unroll
    for (int t = 0; t < 4; ++t) {
        const int n = wave * 64 + t * 16 + l16;
        const float g  = gamma[n];
        const float bt = beta[n];
#pragma unroll
        for (int rt = 0; rt < 2; ++rt) {
#pragma unroll
            for (int r = 0; r < 8; ++r) {
                const int row = rt * 16 + half * 8 + r;
                hn[(size_t)(rowBase + row) * HID + n] =
                    (acc[rt][t][r] - mu[rt][r]) * rs[rt][r] * g + bt;
            }
        }
    }
}

// ---------------------------------------------------------------------------
// Kernel 2: z = h @ w2 + b2 ; zn = z / max(||z||, eps)
// One block = 16 rows; wave w owns cols [w*16, w*16+16) (1 WMMA tile, N=128).
// ---------------------------------------------------------------------------
__global__ __launch_bounds__(256) void k_mlp2(
    const float* __restrict__ hn, const float* __restrict__ w2,
    const float* __restrict__ b2, float* __restrict__ z,
    float* __restrict__ zn)
{
    __shared__ float sh[16][OUT_D];
    __shared__ float psum[16][16];
    __shared__ float srinv[16];

    const int tid  = threadIdx.x;
    const int wave = tid >> 5;
    const int lane = tid & 31;
    const int half = lane >> 4;
    const int l16  = lane & 15;
    const int rowBase = blockIdx.x * 16;
    const int n = wave * 16 + l16;

    const float* hrow = hn + (size_t)(rowBase + l16) * HID;

    v8f acc = v8f{0.f};
    for (int k0 = 0; k0 < HID; k0 += 4) {
        const int ka = k0 + half * 2;
        const v2f a = *(const v2f*)(hrow + ka);
        v2f b;
        b.x = w2[(size_t)ka * OUT_D + n];
        b.y = w2[(size_t)(ka + 1) * OUT_D + n];
        acc = wmma_f32(a, b, acc);
    }

    const float bias = b2[n];
#pragma unroll
    for (int r = 0; r < 8; ++r) {
        const int row = r + half * 8;
        const float v = acc[r] + bias;
        sh[row][n] = v;
        z[(size_t)(rowBase + row) * OUT_D + n] = v;
    }
    __syncthreads();
    {
        const int row = tid >> 4, p = tid & 15;
        float ss = 0.f;
#pragma unroll
        for (int j = 0; j < OUT_D / 16; ++j) {
            const float v = sh[row][p + 16 * j];
            ss += v * v;
        }
        psum[row][p] = ss;
    }
    __syncthreads();
    if (tid < 16) {
        float ss = 0.f;
        for (int p = 0; p < 16; ++p) ss += psum[tid][p];
        srinv[tid] = 1.0f / fmaxf(sqrtf(ss), EPS_COS);
    }
    __syncthreads();
    {
        const int row = tid >> 4, p = tid & 15;
        const float rinv = srinv[row];
        float* out = zn + (size_t)(rowBase + row) * OUT_D;
#pragma unroll
        for (int j = 0; j < OUT_D / 16; ++j) {
            const int c = p + 16 * j;
            out[c] = sh[row][c] * rinv;
        }
    }
}

// ---------------------------------------------------------------------------
// Kernel 3: y_hat = zn @ zn^T   [8192 x 8192], K = 128
// Block = 128x128 tile (grid 64x64); wave w owns a 16x128 row strip.
// The B-strip (zn columns, i.e. transposed rows) is staged in LDS once per
// block in two 64-column passes (~33 KB, stride 132 floats -> the half-wave
// ds_load_b64 pattern covers 64 distinct banks, conflict-free). Inner loop:
// 1 global b64 (A) + 4 ds_load_b64 (B) per 4 WMMAs. This cuts the 8x
// redundant per-wave L2 re-read of the B strip.
// ---------------------------------------------------------------------------
#define GPAD 132

__global__ __launch_bounds__(256) void k_gram(
    const float* __restrict__ zn, float* __restrict__ y)
{
    __shared__ float bsh[64 * GPAD];         // ~33 KB

    const int tid  = threadIdx.x;
    const int wave = tid >> 5;
    const int lane = tid & 31;
    const int half = lane >> 4;
    const int l16  = lane & 15;
    const int rowBase = blockIdx.y * 128 + wave * 16;
    const int colBase = blockIdx.x * 128;

    const float* arow = zn + (size_t)(rowBase + l16) * OUT_D;

    v8f acc[8] = {v8f{0.f}, v8f{0.f}, v8f{0.f}, v8f{0.f},
                  v8f{0.f}, v8f{0.f}, v8f{0.f}, v8f{0.f}};

#pragma unroll
    for (int p = 0; p < 2; ++p) {
        __syncthreads();                     // protect LDS reuse across passes
        // Stage 64 B-rows: thread -> row (tid>>2), quarter-row ((tid&3)*32)
        {
            const int n  = tid >> 2;
            const int kh = (tid & 3) * 32;
            const float4* src =
                (const float4*)(zn + (size_t)(colBase + p * 64 + n) * OUT_D + kh);
            float4* dst = (float4*)(bsh + n * GPAD + kh);
#pragma unroll
            for (int j = 0; j < 8; ++j) dst[j] = src[j];
        }
        __syncthreads();

        for (int k0 = 0; k0 < OUT_D; k0 += 4) {
            const int ka = k0 + half * 2;
            const v2f a = *(const v2f*)(arow + ka);
#pragma unroll
            for (int tt = 0; tt < 4; ++tt) {
                const v2f b = *(const v2f*)(bsh + (tt * 16 + l16) * GPAD + ka);
                acc[p * 4 + tt] = wmma_f32(a, b, acc[p * 4 + tt]);
            }
        }
    }

#pragma unroll
    for (int t = 0; t < 8; ++t) {
        const int col = colBase + t * 16 + l16;
#pragma unroll
        for (int r = 0; r < 8; ++r) {
            const int row = rowBase + r + half * 8;
            y[(size_t)row * B_ROWS + col] = acc[t][r];
        }
    }
}

// ---------------------------------------------------------------------------
extern "C" void kernel_launch(void* const* d_in, const int* in_sizes, int n_in,
                              void* d_out, int out_size, void* d_ws, size_t ws_size,
                              hipStream_t stream) {
    (void)in_sizes; (void)n_in; (void)out_size; (void)ws_size;
    const float* x     = (const float*)d_in[0];
    const float* w1    = (const float*)d_in[1];
    const float* b1    = (const float*)d_in[2];
    const float* pa    = (const float*)d_in[3];
    const float* gamma = (const float*)d_in[4];
    const float* beta  = (const float*)d_in[5];
    const float* w2    = (const float*)d_in[6];
    const float* b2    = (const float*)d_in[7];

    float* out = (float*)d_out;
    float* z   = out;                               // [8192 x 128]
    float* y   = out + (size_t)B_ROWS * OUT_D;      // [8192 x 8192]

    float* hn = (float*)d_ws;                       // [8192 x 512]  16 MB
    float* zn = hn + (size_t)B_ROWS * HID;          // [8192 x 128]   4 MB

    k_mlp1<<<B_ROWS / 32, 256, 0, stream>>>(x, w1, b1, pa, gamma, beta, hn);
    k_mlp2<<<B_ROWS / 16, 256, 0, stream>>>(hn, w2, b2, z, zn);
    k_gram<<<dim3(64, 64), 256, 0, stream>>>(zn, y);
}